// HunYuanTopKGate_91130616086603
// MI455X (gfx1250) — compile-verified
//
#include <hip/hip_runtime.h>
#include <hip/hip_bf16.h>

typedef __attribute__((ext_vector_type(2))) float v2f;
typedef __attribute__((ext_vector_type(4))) float v4f;
typedef __attribute__((ext_vector_type(8))) float v8f;

#define S_TOKENS 4096
#define HID 2048
#define NEXP 16
#define CAP 512
#define CWELEMS ((size_t)S_TOKENS * NEXP * CAP)   // 33554432

// ---------------------------------------------------------------------------
// Kernel 1: logits = x @ wg.T via V_WMMA_F32_16X16X4_F32 (full fp32 precision)
// One wave computes a 16(token) x 16(expert) tile, K-loop over 2048 in steps of 4.
// A: 16x4 fp32, lanes 0-15 hold M rows (VGPR0=K0,VGPR1=K1), lanes 16-31 K2/K3.
// B: 4x16 fp32, lane = column N (=expert), same K split -> contiguous 8B loads
// from wg since B[k][n] = wg[n][k].
// ---------------------------------------------------------------------------
__global__ void logits_wmma(const float* __restrict__ x,
                            const float* __restrict__ wg,
                            float* __restrict__ logits) {
    const int lane = threadIdx.x & 31;
    const int wave = threadIdx.x >> 5;
    const int tile = blockIdx.x * 8 + wave;        // 256 tiles of 16 tokens
    const int m0   = tile * 16;
    const int row  = lane & 15;
    const int kq   = lane >> 4;                    // 0: K=k0,k0+1  1: K=k0+2,k0+3

    const float* ap = x  + (size_t)(m0 + row) * HID + 2 * kq;
    const float* bp = wg + (size_t)row * HID + 2 * kq;

    v8f acc = {};
    #pragma unroll 4
    for (int k0 = 0; k0 < HID; k0 += 4) {
        v2f a = *(const v2f*)(ap + k0);
        v2f b = *(const v2f*)(bp + k0);
        acc = __builtin_amdgcn_wmma_f32_16x16x4_f32(
            /*neg_a=*/false, a, /*neg_b=*/false, b,
            /*c_mod=*/(short)0, acc, /*reuse_a=*/false, /*reuse_b=*/false);
    }
    // D layout: VGPR r -> M = r + 8*(lane>=16), N = lane&15
    float* outp = logits + (size_t)(m0 + 8 * kq) * NEXP + row;
    #pragma unroll
    for (int r = 0; r < 8; ++r) outp[(size_t)r * NEXP] = acc[r];
}

// ---------------------------------------------------------------------------
// Kernel 2: per-token softmax + top-2 (lowest-index tie-break like lax.top_k),
// renormalized selected weights, and per-expert gate sums for l_aux.
// ---------------------------------------------------------------------------
__global__ void gate_topk(const float* __restrict__ logits,
                          int* __restrict__ idx,      // [2][4096]
                          float* __restrict__ wsel,   // [2][4096]
                          float* __restrict__ gate_sum) {
    __shared__ float gsum_sh[NEXP];
    const int t = blockIdx.x * blockDim.x + threadIdx.x;   // token id
    if (threadIdx.x < NEXP) gsum_sh[threadIdx.x] = 0.f;
    __syncthreads();

    float l[NEXP];
    const v4f* lp = (const v4f*)(logits + (size_t)t * NEXP);
    #pragma unroll
    for (int q = 0; q < 4; ++q) {
        v4f v = lp[q];
        l[q*4+0] = v.x; l[q*4+1] = v.y; l[q*4+2] = v.z; l[q*4+3] = v.w;
    }
    float m = l[0];
    #pragma unroll
    for (int e = 1; e < NEXP; ++e) m = fmaxf(m, l[e]);
    float g[NEXP]; float Z = 0.f;
    #pragma unroll
    for (int e = 0; e < NEXP; ++e) { g[e] = __expf(l[e] - m); Z += g[e]; }
    const float inv = 1.0f / Z;

    int i0 = 0; float g0 = g[0];
    #pragma unroll
    for (int e = 1; e < NEXP; ++e) if (g[e] > g0) { g0 = g[e]; i0 = e; }
    int i1 = -1; float g1 = -1e30f;
    #pragma unroll
    for (int e = 0; e < NEXP; ++e) if (e != i0 && g[e] > g1) { g1 = g[e]; i1 = e; }

    g0 *= inv; g1 *= inv;                       // actual softmax gates
    float gs = fmaxf(g0 + g1, 1.1920929e-7f);   // clip by float eps
    idx[t]            = i0;
    idx[S_TOKENS + t] = i1;
    wsel[t]            = g0 / gs;
    wsel[S_TOKENS + t] = g1 / gs;

    #pragma unroll
    for (int e = 0; e < NEXP; ++e) atomicAdd(&gsum_sh[e], g[e] * inv);
    __syncthreads();
    if (threadIdx.x < NEXP) atomicAdd(&gate_sum[threadIdx.x], gsum_sh[threadIdx.x]);
}

// ---------------------------------------------------------------------------
// Kernel 3: rank-major token-ordered scan -> slot priority per selection.
// Single block, 256 threads x 16 tokens each; LDS histogram [256][32] over
// (rank, expert) columns; rank-1 priorities offset by total rank-0 counts.
// ---------------------------------------------------------------------------
__global__ void scan_prio(const int* __restrict__ idx,
                          int* __restrict__ prio,        // [2][4096]
                          int* __restrict__ exp_counts,  // [16]
                          int* __restrict__ validcnt) {
    __shared__ int hist[256][32];
    __shared__ int totals[32];
    __shared__ int validsh;
    const int t = threadIdx.x;
    #pragma unroll
    for (int c = 0; c < 32; ++c) hist[t][c] = 0;
    if (t == 0) validsh = 0;
    const int base = t * 16;
    for (int r = 0; r < 2; ++r)
        for (int j = 0; j < 16; ++j)
            hist[t][r * NEXP + idx[r * S_TOKENS + base + j]]++;
    __syncthreads();
    if (t < 32) {               // serial exclusive scan of one (rank,expert) column
        int run = 0;
        for (int i = 0; i < 256; ++i) { int v = hist[i][t]; hist[i][t] = run; run += v; }
        totals[t] = run;
    }
    __syncthreads();
    int myvalid = 0;
    for (int r = 0; r < 2; ++r) {
        int lc[NEXP];
        #pragma unroll
        for (int e = 0; e < NEXP; ++e) lc[e] = 0;
        for (int j = 0; j < 16; ++j) {
            int tok = base + j;
            int e = idx[r * S_TOKENS + tok];
            int p = hist[t][r * NEXP + e] + lc[e];
            if (r == 1) p += totals[e];          // all rank-0 slots come first
            lc[e]++;
            prio[r * S_TOKENS + tok] = p;
            if (p < CAP) myvalid++;
        }
    }
    atomicAdd(&validsh, myvalid);
    __syncthreads();
    if (t < NEXP) exp_counts[t] = totals[t] + totals[NEXP + t];
    if (t == 0) *validcnt = validsh;
}

// ---------------------------------------------------------------------------
// Kernel 4: scalars + exp_counts tail.
// l_aux = E^2 * mean_e(tpe*rppe) = 16 * sum_e (cnt/4096)*(gsum/4096)
// ---------------------------------------------------------------------------
__global__ void finalize(const float* __restrict__ gate_sum,
                         const int* __restrict__ exp_counts,
                         const int* __restrict__ validcnt,
                         float* __restrict__ out) {
    __shared__ float partial[NEXP];
    const int t = threadIdx.x;
    if (t < NEXP) {
        float tpe  = exp_counts[t] * (1.0f / S_TOKENS);
        float rppe = gate_sum[t]   * (1.0f / S_TOKENS);
        partial[t] = tpe * rppe;
        out[2 + 2 * CWELEMS + t] = (float)exp_counts[t];  // exp_counts tail
    }
    __syncthreads();
    if (t == 0) {
        float s = 0.f;
        for (int e = 0; e < NEXP; ++e) s += partial[e];
        out[0] = 16.0f * s;                                // l_aux
        out[1] = (float)(*validcnt) * (1.0f / (S_TOKENS * 2)); // capacity rate
    }
}

// ---------------------------------------------------------------------------
// Kernel 5: materialize combine_weights + dispatch_mask (268 MB, the roofline
// term: ~11.5us at 23.3 TB/s). One wave per (token, expert) row of 512 floats;
// each lane emits 4x b128 NON-TEMPORAL stores per tensor, nonzero slot fused in.
// ---------------------------------------------------------------------------
__global__ void fill_out(const int* __restrict__ idx,
                         const float* __restrict__ wsel,
                         const int* __restrict__ prio,
                         float* __restrict__ out) {
    const int lane  = threadIdx.x & 31;
    const int wave  = threadIdx.x >> 5;
    const int rowid = blockIdx.x * 8 + wave;   // 0..65535 = token*16 + expert
    const int t = rowid >> 4;
    const int e = rowid & 15;

    int p = -1; float w = 0.f;
    const int i0 = idx[t], i1 = idx[S_TOKENS + t];
    if (i0 == e)      { p = prio[t];            w = wsel[t]; }
    else if (i1 == e) { p = prio[S_TOKENS + t]; w = wsel[S_TOKENS + t]; }
    const bool valid = (p >= 0) && (p < CAP);

    float* cw = out + 2 + (size_t)rowid * CAP;
    float* dm = out + 2 + CWELEMS + (size_t)rowid * CAP;
    #pragma unroll
    for (int i = 0; i < 4; ++i) {
        const int c = i * 128 + lane * 4;      // wave writes 512B contiguous/iter
        v4f cwv = {0.f, 0.f, 0.f, 0.f};
        v4f dmv = {0.f, 0.f, 0.f, 0.f};
        if (valid && ((p >> 2) == (c >> 2))) { cwv[p & 3] = w; dmv[p & 3] = 1.0f; }
        __builtin_nontemporal_store(cwv, (v4f*)(cw + c));
        __builtin_nontemporal_store(dmv, (v4f*)(dm + c));
    }
}

extern "C" void kernel_launch(void* const* d_in, const int* in_sizes, int n_in,
                              void* d_out, int out_size, void* d_ws, size_t ws_size,
                              hipStream_t stream) {
    const float* x  = (const float*)d_in[0];   // [2,2048,2048] -> [4096,2048]
    const float* wg = (const float*)d_in[1];   // [16,2048]
    float* out = (float*)d_out;
    float* ws  = (float*)d_ws;

    float* logits    = ws;                       // 65536 f32
    int*   idx       = (int*)(ws + 65536);       // [2][4096] i32
    float* wsel      = ws + 65536 + 8192;        // [2][4096] f32
    int*   prio      = (int*)(ws + 81920);       // [2][4096] i32
    float* gate_sum  = ws + 90112;               // [16] f32 (atomic accum)
    int*   expcnt    = (int*)(ws + 90128);       // [16] i32
    int*   validcnt  = (int*)(ws + 90144);       // [1] i32

    hipMemsetAsync(gate_sum, 0, 33 * sizeof(float), stream);

    logits_wmma<<<dim3(32), dim3(256), 0, stream>>>(x, wg, logits);
    gate_topk  <<<dim3(16), dim3(256), 0, stream>>>(logits, idx, wsel, gate_sum);
    scan_prio  <<<dim3(1),  dim3(256), 0, stream>>>(idx, prio, expcnt, validcnt);
    finalize   <<<dim3(1),  dim3(32),  0, stream>>>(gate_sum, expcnt, validcnt, out);
    fill_out   <<<dim3(8192), dim3(256), 0, stream>>>(idx, wsel, prio, out);
}